// QuantumBandit_59004260713060
// MI455X (gfx1250) — compile-verified
//
#include <hip/hip_runtime.h>
#include <math.h>

// ---------------------------------------------------------------------------
// QuantumBandit fused kernel for gfx1250 (MI455X).
//  Phase 1: Z = tanh(feats @ W^T + b) via V_WMMA_F32_16X16X4_F32 (exact fp32).
//  Phase 2: per-sample 6-qubit statevector sim in packed-f32 (V_PK_FMA_F32)
//           complex arithmetic; algebraically reduced: last-layer CZs
//           (diagonal) and last-layer Rots on wires 1..5 (unitary on
//           traced-out wires) dropped -- exact for <Z_0>.
// Memory floor: 268 MB feats / 23.3 TB/s ~= 11.5 us; packed math halves the
// competing VALU cost of the sim phase.
// ---------------------------------------------------------------------------

typedef __attribute__((ext_vector_type(2))) float v2f;
typedef __attribute__((ext_vector_type(8))) float v8f;

#define NQ 6
#define NF 256
#define WROW 260                 // padded LDS row stride (dwords); 260 % 64 == 4 -> conflict-free
#define TPB 128                  // 4 waves per block
#define SAMPLES_PER_BLOCK 128    // 32 samples per wave (two 16-row WMMA tiles)

static __device__ __forceinline__ v2f mk2(float a, float b) {
    v2f r; r.x = a; r.y = b; return r;
}

// branch-free tanh: 1 - 2/(e^{2x}+1); exp overflow/underflow give exactly +-1
static __device__ __forceinline__ float fast_tanh(float x) {
    return 1.0f - 2.0f * __builtin_amdgcn_rcpf(__expf(2.0f * x) + 1.0f);
}

__global__ __launch_bounds__(TPB) void qbandit_kernel(
    const float* __restrict__ feats,   // (B, 256)
    const float* __restrict__ Wpre,    // (6, 256)
    const float* __restrict__ bpre,    // (6,)
    const float* __restrict__ wts,     // (2, 6, 3)
    float* __restrict__ out)           // (B,)
{
    __shared__ float wLds[NQ * WROW];             // padded W_pre rows
    __shared__ v2f   gLds[12 * 8];                // per gate: c00R,c00S,c01R,c01S,c10R,c10S,c11R,c11S
    __shared__ float biasLds[8];
    __shared__ float zLds[SAMPLES_PER_BLOCK * 8]; // per-sample 6 angles, padded to 8

    const int t    = threadIdx.x;
    const int lane = t & 31;
    const int wave = t >> 5;

    // ---- stage W_pre into padded LDS rows (coalesced) ----
    for (int i = t; i < NQ * NF; i += TPB) {
        int n = i >> 8, k = i & 255;
        wLds[n * WROW + k] = Wpre[i];
    }
    if (t < NQ) biasLds[t] = bpre[t];

    // ---- precompute the 12 fixed Rot(phi,theta,omega) gates in packed form.
    // complex mul u*a packs as {ur,ur}*{ar,ai} + {-ui,ui}*{ai,ar}.
    if (t < 12) {
        float phi = wts[t * 3 + 0], th = wts[t * 3 + 1], om = wts[t * 3 + 2];
        float cth = __cosf(0.5f * th), sth = __sinf(0.5f * th);
        float apo = -0.5f * (phi + om);   // u00 phase
        float amo =  0.5f * (phi - om);   // u01 phase
        float u00r =  __cosf(apo) * cth, u00i =  __sinf(apo) * cth;
        float u01r = -__cosf(amo) * sth, u01i = -__sinf(amo) * sth;
        float u10r =  __cosf(amo) * sth, u10i = -__sinf(amo) * sth;
        float u11r =  __cosf(apo) * cth, u11i = -__sinf(apo) * cth;
        v2f* g = &gLds[t * 8];
        g[0] = mk2(u00r, u00r); g[1] = mk2(-u00i, u00i);
        g[2] = mk2(u01r, u01r); g[3] = mk2(-u01i, u01i);
        g[4] = mk2(u10r, u10r); g[5] = mk2(-u10i, u10i);
        g[6] = mk2(u11r, u11r); g[7] = mk2(-u11i, u11i);
    }
    __syncthreads();

    // =====================================================================
    // Phase 1: GEMM tile via V_WMMA_F32_16X16X4_F32.
    //   A (16x4): lane L holds row M=L&15, k = 4*kk + (L>>4)*2 + {0,1}
    //   B (4x16): lane L holds col N=L&15, same k pair  (N>=6 reads n%6, unused)
    //   C (16x16): lane L vgpr r holds D[M = r + 8*(L>>4)][N = L&15]
    // =====================================================================
    const size_t rowBase = (size_t)(blockIdx.x * SAMPLES_PER_BLOCK + wave * 32) * NF;
    const float* f0 = feats + rowBase + (size_t)(lane & 15) * NF;  // tile0: rows 0..15
    const float* f1 = f0 + 16 * NF;                                // tile1: rows 16..31
    const int khalf = (lane >> 4) * 2;
    const float* wrow = &wLds[((lane & 15) % NQ) * WROW];

    v8f acc0 = {};
    v8f acc1 = {};
#pragma unroll 8
    for (int kk = 0; kk < 64; ++kk) {
        const int kb = kk * 4 + khalf;
        v2f a0 = *(const v2f*)(f0 + kb);
        v2f a1 = *(const v2f*)(f1 + kb);
        v2f b  = *(const v2f*)(wrow + kb);
        acc0 = __builtin_amdgcn_wmma_f32_16x16x4_f32(false, a0, false, b,
                                                     (short)0, acc0, false, false);
        acc1 = __builtin_amdgcn_wmma_f32_16x16x4_f32(false, a1, false, b,
                                                     (short)0, acc1, false, false);
    }

    // ---- bias + fast tanh, scatter to per-sample layout in LDS ----
    {
        const int n = lane & 15;
        if (n < NQ) {
            float bv = biasLds[n];
#pragma unroll
            for (int r = 0; r < 8; ++r) {
                int m = (lane >> 4) * 8 + r;   // sample within 16-row tile
                zLds[(wave * 32 + m) * 8 + n]      = fast_tanh(acc0[r] + bv);
                zLds[(wave * 32 + 16 + m) * 8 + n] = fast_tanh(acc1[r] + bv);
            }
        }
    }
    __syncthreads();

    // =====================================================================
    // Phase 2: one sample per lane; state as packed (re,im) pairs so the
    // complex 2x2 gate updates lower to V_PK_FMA_F32 / V_PK_MUL_F32.
    // State index i: bit (5-w) <-> wire w.
    // =====================================================================
    float cs[NQ], sn[NQ];
#pragma unroll
    for (int w = 0; w < NQ; ++w) {
        float h = 0.5f * zLds[t * 8 + w];   // |h| <= 0.5: native trig is safe
        cs[w] = __cosf(h);
        sn[w] = __sinf(h);
    }

    // RX product state |psi> = kron_w [c_w, -i s_w] by iterative doubling.
    // (-i s)*x = {s*xi, -s*xr} = {s,-s} * x.yx  -> one pk_mul per new amp.
    v2f st[64];
    st[0] = mk2(1.0f, 0.0f);
#pragma unroll
    for (int w = 5; w >= 0; --w) {
        const int size = 1 << (5 - w);
        const v2f cw  = mk2(cs[w],  cs[w]);
        const v2f swp = mk2(sn[w], -sn[w]);
#pragma unroll
        for (int a = 0; a < size; ++a) {
            v2f x = st[a];
            st[a + size] = swp * x.yx;
            st[a]        = cw * x;
        }
    }

    // Layer 0: Rot on all 6 wires (packed complex 2x2 updates).
#pragma unroll
    for (int w = 0; w < NQ; ++w) {
        const v2f c00R = gLds[w * 8 + 0], c00S = gLds[w * 8 + 1];
        const v2f c01R = gLds[w * 8 + 2], c01S = gLds[w * 8 + 3];
        const v2f c10R = gLds[w * 8 + 4], c10S = gLds[w * 8 + 5];
        const v2f c11R = gLds[w * 8 + 6], c11S = gLds[w * 8 + 7];
        const int stride = 1 << (5 - w);
#pragma unroll
        for (int base = 0; base < 64; base += 2 * stride) {
#pragma unroll
            for (int off = 0; off < stride; ++off) {
                const int i = base + off, j = i + stride;
                v2f a = st[i], b = st[j];
                v2f as = a.yx, bs = b.yx;
                v2f ni = c00R * a; ni += c00S * as; ni += c01R * b; ni += c01S * bs;
                v2f nj = c10R * a; nj += c10S * as; nj += c11R * b; nj += c11S * bs;
                st[i] = ni; st[j] = nj;
            }
        }
    }

    // Layer 0 CZs on pairs (0,1),(2,3),(4,5),(1,2),(3,4): one compile-time
    // parity-folded sign pass.
#pragma unroll
    for (int i = 0; i < 64; ++i) {
        const int b5 = (i >> 5) & 1, b4 = (i >> 4) & 1, b3 = (i >> 3) & 1;
        const int b2 = (i >> 2) & 1, b1 = (i >> 1) & 1, b0 = i & 1;
        const int par = (b5 & b4) ^ (b3 & b2) ^ (b1 & b0) ^ (b4 & b3) ^ (b2 & b1);
        if (par) st[i] = -st[i];
    }

    // Layer 1: only the wire-0 Rot affects <Z_0>. Gate index 6 = (l=1, w=0).
    {
        const v2f c00R = gLds[6 * 8 + 0], c00S = gLds[6 * 8 + 1];
        const v2f c01R = gLds[6 * 8 + 2], c01S = gLds[6 * 8 + 3];
        const v2f c10R = gLds[6 * 8 + 4], c10S = gLds[6 * 8 + 5];
        const v2f c11R = gLds[6 * 8 + 6], c11S = gLds[6 * 8 + 7];
#pragma unroll
        for (int i = 0; i < 32; ++i) {
            const int j = i + 32;
            v2f a = st[i], b = st[j];
            v2f as = a.yx, bs = b.yx;
            v2f ni = c00R * a; ni += c00S * as; ni += c01R * b; ni += c01S * bs;
            v2f nj = c10R * a; nj += c10S * as; nj += c11R * b; nj += c11S * bs;
            st[i] = ni; st[j] = nj;
        }
    }

    // <Z_0> = P(bit5=0) - P(bit5=1), packed signed accumulation.
    v2f acc = mk2(0.0f, 0.0f);
#pragma unroll
    for (int i = 0; i < 32; ++i) acc += st[i] * st[i];
#pragma unroll
    for (int i = 32; i < 64; ++i) acc -= st[i] * st[i];
    const float logit = acc.x + acc.y;

    out[(size_t)blockIdx.x * SAMPLES_PER_BLOCK + t] =
        __builtin_amdgcn_rcpf(1.0f + __expf(-2.0f * logit));
}

extern "C" void kernel_launch(void* const* d_in, const int* in_sizes, int n_in,
                              void* d_out, int out_size, void* d_ws, size_t ws_size,
                              hipStream_t stream) {
    const float* feats = (const float*)d_in[0];
    const float* Wpre  = (const float*)d_in[1];
    const float* bpre  = (const float*)d_in[2];
    const float* wts   = (const float*)d_in[3];
    float* out = (float*)d_out;

    const int batch  = in_sizes[0] / NF;               // 262144
    const int blocks = batch / SAMPLES_PER_BLOCK;      // 2048
    qbandit_kernel<<<blocks, TPB, 0, stream>>>(feats, Wpre, bpre, wts, out);
}